// SClayer_23545010716732
// MI455X (gfx1250) — compile-verified
//
#include <hip/hip_runtime.h>
#include <hip/hip_bf16.h>
#include <math.h>

// ---------------------------------------------------------------------------
// FISTA sparse coding: T=36, K = 4*161+1 = 645, P=8192, 40 iterations.
// Each iteration: x = A*y + (DtY - lambda); y = (1+tt)*x - tt*x_old
// -> serial chain of [645x645]x[645x8192] f32 GEMMs (~275 GFLOP), L2-resident.
// Matrix pipe: V_WMMA_F32_16X16X4_F32.
// y is stored TRANSPOSED ([P][K]) so both A and B chunks are row-contiguous:
// async global->LDS b128 DMA stages them directly in WMMA-friendly layout
// (zero staging VGPRs, zero operand-shuffle movs in the hot loop).
// ---------------------------------------------------------------------------
#define TDIM    36
#define KCOLS   645          // real matrix dimension
#define MPAD    672          // padded M and K (42*16, 21*32)
#define NCOLS   8192
#define NPOLES  161
#define KC      32           // K chunk staged in LDS
#define NCH     (MPAD / KC)  // 21 chunks
#define WGN     128          // N columns per workgroup (8 tiles x 16)
#define LDS_ST  36           // LDS row stride in floats (32 + 4 pad)

typedef float v2f __attribute__((ext_vector_type(2)));
typedef float v8f __attribute__((ext_vector_type(8)));

#if __has_builtin(__builtin_amdgcn_global_load_async_to_lds_b128)
#define HAVE_ASYNC_LDS 1
typedef int v4i __attribute__((vector_size(4 * sizeof(int))));
__device__ __forceinline__ void async_copy16(const float* g, float* l) {
  __builtin_amdgcn_global_load_async_to_lds_b128((v4i*)g, (v4i*)l, 0, 0);
}
#define WAIT_ASYNC0() asm volatile("s_wait_asynccnt 0" ::: "memory")
#else
#define HAVE_ASYNC_LDS 0
#endif

// ------------------------- dictionary build --------------------------------
__global__ __launch_bounds__(256) void build_dict(const float* __restrict__ rr,
                                                  const float* __restrict__ theta,
                                                  float* __restrict__ D) {
  int j = blockIdx.x * blockDim.x + threadIdx.x;
  if (j >= KCOLS) return;
  float v[TDIM];
  if (j == 0) {
#pragma unroll
    for (int i = 0; i < TDIM; ++i) v[i] = 1.0f;
  } else {
    int g = (j - 1) / NPOLES;        // 0: r^i cos, 1: (-r)^i cos, 2: r^i sin, 3: (-r)^i sin
    int idx = (j - 1) - g * NPOLES;
    float r = rr[idx];
    float th = theta[idx];
    bool alt = (g == 1) || (g == 3);
    bool use_sin = (g >= 2);
    float ri = 1.0f;
#pragma unroll
    for (int i = 0; i < TDIM; ++i) {
      float ang = (float)i * th;
      float tr = use_sin ? __sinf(ang) : __cosf(ang);
      float sgn = (alt && (i & 1)) ? -1.0f : 1.0f;
      v[i] = sgn * ri * tr;
      ri *= r;
    }
  }
  float n2 = 0.0f;
#pragma unroll
  for (int i = 0; i < TDIM; ++i) n2 = fmaf(v[i], v[i], n2);
  float G = sqrtf(n2);
  if (G == 0.0f) G = 6.0f;           // sqrt(T)
  float inv = 1.0f / G;
#pragma unroll
  for (int i = 0; i < TDIM; ++i) D[i * KCOLS + j] = v[i] * inv;
}

// ------------------------- DtD = D^T D -------------------------------------
__global__ __launch_bounds__(256) void build_dtd(const float* __restrict__ D,
                                                 float* __restrict__ DtD) {
  int idx = blockIdx.x * blockDim.x + threadIdx.x;
  if (idx >= KCOLS * KCOLS) return;
  int r = idx / KCOLS;
  int c = idx - r * KCOLS;
  float s = 0.0f;
#pragma unroll
  for (int t = 0; t < TDIM; ++t) s = fmaf(D[t * KCOLS + r], D[t * KCOLS + c], s);
  DtD[idx] = s;
}

// ---------------- Frobenius norm of DtD -> linv, lambda (deterministic) ----
__global__ __launch_bounds__(1024) void reduce_scalars(const float* __restrict__ DtD,
                                                       float* __restrict__ scal) {
  __shared__ float red[1024];
  float s = 0.0f;
  for (int i = threadIdx.x; i < KCOLS * KCOLS; i += 1024) {
    float v = DtD[i];
    s = fmaf(v, v, s);
  }
  red[threadIdx.x] = s;
  __syncthreads();
  for (int off = 512; off > 0; off >>= 1) {
    if ((int)threadIdx.x < off) red[threadIdx.x] += red[threadIdx.x + off];
    __syncthreads();
  }
  if (threadIdx.x == 0) {
    float L = sqrtf(red[0]);
    float linv = (L == 0.0f) ? 0.0f : 1.0f / L;
    scal[0] = linv;
    scal[1] = 0.05f * linv;          // GAMMA * linv
  }
}

// ---------------- A = I - DtD*linv, zero-padded to MPAD x MPAD --------------
__global__ __launch_bounds__(256) void build_A(const float* __restrict__ DtD,
                                               const float* __restrict__ scal,
                                               float* __restrict__ Apad) {
  int idx = blockIdx.x * blockDim.x + threadIdx.x;
  if (idx >= MPAD * MPAD) return;
  int r = idx / MPAD;
  int c = idx - r * MPAD;
  float v = 0.0f;
  if (r < KCOLS && c < KCOLS)
    v = ((r == c) ? 1.0f : 0.0f) - scal[0] * DtD[r * KCOLS + c];
  Apad[idx] = v;
}

// ---------------- DtY = (D^T Y) * linv  [645 x 8192] -----------------------
__global__ __launch_bounds__(256) void build_dty(const float* __restrict__ D,
                                                 const float* __restrict__ Y,
                                                 const float* __restrict__ scal,
                                                 float* __restrict__ DtY) {
  int idx = blockIdx.x * blockDim.x + threadIdx.x;
  if (idx >= KCOLS * NCOLS) return;
  int k = idx >> 13;                 // / 8192
  int p = idx & (NCOLS - 1);
  float s = 0.0f;
#pragma unroll
  for (int t = 0; t < TDIM; ++t) s = fmaf(D[t * KCOLS + k], Y[t * NCOLS + p], s);
  DtY[idx] = s * scal[0];
}

// ---------------- main fused FISTA step: Z = A*y; x,y update ----------------
// grid = (8192/128, 6), block = 256 (8 wave32s). WG tile: 128(M) x 128(N).
// Each wave: one 16-row M tile, eight 16x16 f32 WMMA accumulators (16x128).
// yinT/youtT are [P][K]-transposed so A and B chunks are both row-contiguous.
// LDS double-buffered; chunk i+1 copied global->LDS by async DMA while chunk
// i runs on the matrix pipe. B fragments: single aligned ds_load_b64 each.
__global__ __launch_bounds__(256) __attribute__((amdgpu_waves_per_eu(1)))
void fista_step(const float* __restrict__ Apad,
                const float* __restrict__ yinT,
                const float* __restrict__ DtY,
                const float* __restrict__ scal,
                float* __restrict__ x,
                float* __restrict__ youtT,
                float tt) {
  __shared__ float ldsA[2][128 * LDS_ST];    // 2 x 18432 B: A rows, [m][k]
  __shared__ float ldsBT[2][WGN * LDS_ST];   // 2 x 18432 B: yT rows, [n][k]

  const int tid = threadIdx.x;
  const int lane = tid & 31;
  const int wave = tid >> 5;              // 0..7
  const int n0 = blockIdx.x * WGN;
  const int mbase = blockIdx.y * 128;
  int mt = blockIdx.y * 8 + wave;         // M tile index
  if (mt > (MPAD / 16 - 1)) mt = MPAD / 16 - 1;   // clamp (band-5 overhang)
  const int mloc = mt * 16 - mbase;       // wave's row base inside LDS A tile
  const int mrow = lane & 15;             // A row within tile
  const int khalf = lane >> 4;            // half-wave selector (K pair / C row half)
  const int ncol = lane & 15;             // B/C column within tile

  // staging split (identical pattern for A and BT): 128 rows x 32 k,
  // 4 x b128 per thread for each of the two tiles
  const int sRow = tid >> 1;              // 0..127
  const int sSeg = (tid & 1) * 16;        // 0 or 16
  int sAgrow = mbase + sRow;
  if (sAgrow > MPAD - 1) sAgrow = MPAD - 1;  // clamped rows are never consumed

  v8f acc[8];
#pragma unroll
  for (int t = 0; t < 8; ++t) acc[t] = (v8f)(0.0f);

#if HAVE_ASYNC_LDS
  auto issue_async = [&](int k0, int buf) {
    const float* ag = Apad + sAgrow * MPAD + k0 + sSeg;
    float* al = &ldsA[buf][sRow * LDS_ST + sSeg];
#pragma unroll
    for (int i = 0; i < 4; ++i) async_copy16(ag + 4 * i, al + 4 * i);
    const float* bg = yinT + (size_t)(n0 + sRow) * MPAD + k0 + sSeg;
    float* bl = &ldsBT[buf][sRow * LDS_ST + sSeg];
#pragma unroll
    for (int i = 0; i < 4; ++i) async_copy16(bg + 4 * i, bl + 4 * i);
  };
  issue_async(0, 0);
#else
  float4 Ar[4], Br[4];
  auto load_global = [&](int k0) {
    const float* ag = Apad + sAgrow * MPAD + k0 + sSeg;
    const float* bg = yinT + (size_t)(n0 + sRow) * MPAD + k0 + sSeg;
#pragma unroll
    for (int i = 0; i < 4; ++i) { Ar[i] = *(const float4*)(ag + 4 * i);
                                  Br[i] = *(const float4*)(bg + 4 * i); }
  };
  auto store_lds = [&](int buf) {
    float4* ad = (float4*)&ldsA[buf][sRow * LDS_ST + sSeg];
    float4* bd = (float4*)&ldsBT[buf][sRow * LDS_ST + sSeg];
#pragma unroll
    for (int i = 0; i < 4; ++i) { ad[i] = Ar[i]; bd[i] = Br[i]; }
  };
  load_global(0);
  store_lds(0);
#endif

  for (int ic = 0; ic < NCH; ++ic) {
#if HAVE_ASYNC_LDS
    WAIT_ASYNC0();                        // this wave's copies for chunk ic done
#endif
    __syncthreads();                      // all waves' copies done; prev reads done
    const int cur = ic & 1;
#if HAVE_ASYNC_LDS
    if (ic + 1 < NCH) issue_async((ic + 1) * KC, cur ^ 1);  // overlaps WMMA below
#else
    if (ic + 1 < NCH) load_global((ic + 1) * KC);
#endif

    const float* arow = &ldsA[cur][(mloc + mrow) * LDS_ST + 2 * khalf];
    const float* bcol = &ldsBT[cur][ncol * LDS_ST + 2 * khalf];
#pragma unroll
    for (int kk = 0; kk < KC; kk += 4) {
      v2f a = *(const v2f*)(arow + kk);               // A[m, kk+2h .. +1]
#pragma unroll
      for (int t = 0; t < 8; ++t) {
        v2f b = *(const v2f*)(bcol + t * (16 * LDS_ST) + kk);  // yT[n+16t, kk+2h..+1]
        acc[t] = __builtin_amdgcn_wmma_f32_16x16x4_f32(false, a, false, b,
                                                       (short)0, acc[t], false, false);
      }
    }

#if !HAVE_ASYNC_LDS
    if (ic + 1 < NCH) store_lds(cur ^ 1);
#endif
  }

  // ---- fused FISTA epilogue -------------------------------------------------
  // x, DtY are row-major [K][P] (lane-coalesced); y output is transposed:
  // a lane's 8 accumulator rows are 8 consecutive m -> contiguous 32B of yT[p].
  const float lambd = scal[1];
  const float om = 1.0f + tt;
  const int mchunk = mt * 16 + 8 * khalf;   // base row of this lane's C chunk
#pragma unroll
  for (int t = 0; t < 8; ++t) {
    const int p = n0 + t * 16 + ncol;
    if (mchunk + 7 < KCOLS) {               // fast path: fully in range
      float yv[8];
#pragma unroll
      for (int r = 0; r < 8; ++r) {
        const int idx = (mchunk + r) * NCOLS + p;
        const float xn = acc[t][r] + DtY[idx] - lambd;
        const float xo = x[idx];
        x[idx] = xn;
        yv[r] = om * xn - tt * xo;
      }
      float4* yd = (float4*)&youtT[(size_t)p * MPAD + mchunk];
      yd[0] = float4{yv[0], yv[1], yv[2], yv[3]};
      yd[1] = float4{yv[4], yv[5], yv[6], yv[7]};
    } else {                                // boundary: per-element masking
#pragma unroll
      for (int r = 0; r < 8; ++r) {
        const int m = mchunk + r;
        if (m < KCOLS) {
          const int idx = m * NCOLS + p;
          const float xn = acc[t][r] + DtY[idx] - lambd;
          const float xo = x[idx];
          x[idx] = xn;
          youtT[(size_t)p * MPAD + m] = om * xn - tt * xo;
        }
      }
    }
  }
}

// ---------------------------------------------------------------------------
extern "C" void kernel_launch(void* const* d_in, const int* in_sizes, int n_in,
                              void* d_out, int out_size, void* d_ws, size_t ws_size,
                              hipStream_t stream) {
  (void)in_sizes; (void)n_in; (void)out_size; (void)ws_size;
  const float* Y     = (const float*)d_in[0];   // [1,36,8192]
  const float* rr    = (const float*)d_in[1];   // [161]
  const float* theta = (const float*)d_in[2];   // [161]
  float* x = (float*)d_out;                     // [645,8192] == x buffer (in place)

  // workspace layout (floats, 16-float aligned)
  float* ws   = (float*)d_ws;
  float* Apad = ws;                                  // 672*672      = 451584
  float* D    = Apad + (size_t)MPAD * MPAD;          // 36*645 -> pad 23232
  float* DtD  = D + 23232;                           // 645*645 -> pad 416032
  float* scal = DtD + 416032;                        // 16
  float* DtY  = scal + 16;                           // 645*8192     = 5283840
  float* y0   = DtY + (size_t)KCOLS * NCOLS;         // 8192*672     = 5505024 (yT)
  float* y1   = y0 + (size_t)NCOLS * MPAD;           // total ~68.7 MB

  // momentum coefficients (data-independent; fixed sequence of 40 launches)
  float tts[40];
  double t = 1.0;
  for (int i = 0; i < 40; ++i) {
    double tn = (1.0 + sqrt(1.0 + 4.0 * t * t)) / 2.0;
    tts[i] = (float)((t - 1.0) / tn);
    t = tn;
  }

  // zero-init: yT ping-pong buffers (incl. K padding cols) and x (= d_out)
  (void)hipMemsetAsync(y0, 0, (size_t)NCOLS * MPAD * sizeof(float), stream);
  (void)hipMemsetAsync(y1, 0, (size_t)NCOLS * MPAD * sizeof(float), stream);
  (void)hipMemsetAsync(x, 0, (size_t)KCOLS * NCOLS * sizeof(float), stream);

  build_dict<<<(KCOLS + 255) / 256, 256, 0, stream>>>(rr, theta, D);
  build_dtd<<<(KCOLS * KCOLS + 255) / 256, 256, 0, stream>>>(D, DtD);
  reduce_scalars<<<1, 1024, 0, stream>>>(DtD, scal);
  build_A<<<(MPAD * MPAD + 255) / 256, 256, 0, stream>>>(DtD, scal, Apad);
  build_dty<<<(KCOLS * NCOLS + 255) / 256, 256, 0, stream>>>(D, Y, scal, DtY);

  dim3 grid(NCOLS / WGN, 6);   // 64 N-blocks x 6 M-bands (42 tiles of 16 rows)
  for (int i = 0; i < 40; ++i) {
    const float* yi = (i & 1) ? y1 : y0;
    float*       yo = (i & 1) ? y0 : y1;
    fista_step<<<grid, 256, 0, stream>>>(Apad, yi, DtY, scal, x, yo, tts[i]);
  }
  // x (d_out) now holds x_final rows 0..644 — exactly the reference output.
}